// QLSTM_65481071401092
// MI455X (gfx1250) — compile-verified
//
#include <hip/hip_runtime.h>

typedef float v2f __attribute__((ext_vector_type(2)));
typedef float v8f __attribute__((ext_vector_type(8)));

#define S_LEN 256
#define BATCH 8192
#define HID   8
#define CSTR  36   // comb row stride (dwords): conflict-free, 16B-aligned rows
#define GSTR  36   // gate row stride (dwords): conflict-free, 16B-aligned quads

__device__ __forceinline__ float fsigmoid(float v) {
  return __builtin_amdgcn_rcpf(1.0f + __expf(-v));
}
__device__ __forceinline__ float ftanh(float v) {
  return 1.0f - 2.0f * __builtin_amdgcn_rcpf(__expf(2.0f * v) + 1.0f);
}

__global__ __launch_bounds__(32) void qlstm_scan_kernel(
    const float* __restrict__ X,        // (S,B,16)
    const float* __restrict__ conv_w,   // (16,16,3)
    const float* __restrict__ conv_b,
    const float* __restrict__ bn_gamma,
    const float* __restrict__ bn_beta,
    const float* __restrict__ bn_mean,
    const float* __restrict__ bn_var,
    const float* __restrict__ w_f, const float* __restrict__ b_f,
    const float* __restrict__ w_i, const float* __restrict__ b_i,
    const float* __restrict__ w_g, const float* __restrict__ b_g,
    const float* __restrict__ w_o, const float* __restrict__ b_o,
    const float* __restrict__ q_params, // (8,3)
    float* __restrict__ out)            // outputs (S,B,8) | hx (B,8) | cx (B,8)
{
  __shared__ float comb[16 * CSTR];   // cols 0..15: x_enc, 16..23: hx
  __shared__ float gate[16 * GSTR];   // cols 0..15: f|i, 16..31: g|o

  const int lane = threadIdx.x;       // 0..31 (wave32)
  const int row  = lane & 15;         // A-row / C-column index
  const int hi   = lane >> 4;         // half-wave select
  const int h4   = hi << 2;           // hidden-quad base for elementwise phase
  const int b0   = blockIdx.x << 4;   // 16-sample batch tile

  // ---- conv weights as WMMA B fragments: wc_mid^T, 4 K-chunks of 4 ----
  v2f bwc[4];
  #pragma unroll
  for (int k = 0; k < 4; ++k) {
    const int d0 = 4 * k + 2 * hi;
    bwc[k].x = conv_w[row * 48 + (d0 + 0) * 3 + 1];   // middle tap only
    bwc[k].y = conv_w[row * 48 + (d0 + 1) * 3 + 1];
  }

  // ---- fused gate weights W_all^T (24x32): N-tile0 = f|i, N-tile1 = g|o ----
  const float* Wt0 = (row < 8) ? (w_f + row * 24) : (w_i + (row - 8) * 24);
  const float* Wt1 = (row < 8) ? (w_g + row * 24) : (w_o + (row - 8) * 24);
  v2f bwg0[6], bwg1[6];
  #pragma unroll
  for (int k = 0; k < 6; ++k) {
    const int p0 = 4 * k + 2 * hi;
    bwg0[k].x = Wt0[p0]; bwg0[k].y = Wt0[p0 + 1];
    bwg1[k].x = Wt1[p0]; bwg1[k].y = Wt1[p0 + 1];
  }
  const float gb0 = (row < 8) ? b_f[row] : b_i[row - 8]; // bias seeded into C
  const float gb1 = (row < 8) ? b_g[row] : b_o[row - 8];

  // BN constants for this lane's channel (c = row); conv bias folded in
  const float bscale = bn_gamma[row] * rsqrtf(bn_var[row] + 1e-5f);
  const float bshift = bn_beta[row];
  const float bmean  = bn_mean[row] - conv_b[row];

  // Closed-form quantum constants: z_k = cos(b_k) * cos(x_enc_k + a_k)
  float qa[8], qcb[8];
  #pragma unroll
  for (int k = 0; k < 8; ++k) {
    qa[k]  = q_params[3 * k + 0];
    qcb[k] = __cosf(q_params[3 * k + 1]);
  }

  float cx[4] = {0.f, 0.f, 0.f, 0.f}; // cell state for (m=row, h=h4+j)

  if (hi == 0) {                       // hx(0) = 0
    #pragma unroll
    for (int j = 0; j < 8; ++j) comb[row * CSTR + 16 + j] = 0.0f;
  }
  __syncthreads();

  const float* xlane = X + ((size_t)b0 + row) * 16 + 2 * hi;

  for (int s = 0; s < S_LEN; ++s) {
    const float* xs = xlane + (size_t)s * (BATCH * 16);
    if (s + 1 < S_LEN) __builtin_prefetch(xs + BATCH * 16, 0, 1);

    // ---- conv GEMM: y(16x16) = x(16x16) @ wc_mid^T, K split 4x4 ----
    v8f y = {0.f,0.f,0.f,0.f,0.f,0.f,0.f,0.f};
    #pragma unroll
    for (int k = 0; k < 4; ++k) {
      v2f a = *(const v2f*)(xs + 4 * k);
      y = __builtin_amdgcn_wmma_f32_16x16x4_f32(false, a, false, bwc[k],
                                                (short)0, y, false, false);
    }

    // ---- BN + ReLU; lane owns channel c=row, rows v+8*hi ----
    #pragma unroll
    for (int v = 0; v < 8; ++v) {
      float e = (y[v] - bmean) * bscale + bshift;
      comb[(v + 8 * hi) * CSTR + row] = fmaxf(e, 0.0f);
    }
    __syncthreads();

    // ---- gate GEMM: (16x24) @ (24x32) as 6 K-chunks x 2 N-tiles ----
    v8f c1 = {gb0,gb0,gb0,gb0,gb0,gb0,gb0,gb0};
    v8f c2 = {gb1,gb1,gb1,gb1,gb1,gb1,gb1,gb1};
    #pragma unroll
    for (int k = 0; k < 6; ++k) {
      v2f a = *(const v2f*)(&comb[row * CSTR + 4 * k + 2 * hi]);
      c1 = __builtin_amdgcn_wmma_f32_16x16x4_f32(false, a, false, bwg0[k],
                                                 (short)0, c1, false, false);
      c2 = __builtin_amdgcn_wmma_f32_16x16x4_f32(false, a, false, bwg1[k],
                                                 (short)0, c2, false, false);
    }
    #pragma unroll
    for (int v = 0; v < 8; ++v) {
      gate[(v + 8 * hi) * GSTR + row]      = c1[v];
      gate[(v + 8 * hi) * GSTR + 16 + row] = c2[v];
    }
    __syncthreads();

    // ---- elementwise: lane handles sample m=row, hidden h = h4 + j ----
    // q via closed form: vectorized b128 reads of x_enc[0..7]
    const float4 xq0 = *(const float4*)&comb[row * CSTR + 0];
    const float4 xq1 = *(const float4*)&comb[row * CSTR + 4];
    float z[8];
    z[0] = qcb[0] * __cosf(xq0.x + qa[0]);
    z[1] = qcb[1] * __cosf(xq0.y + qa[1]);
    z[2] = qcb[2] * __cosf(xq0.z + qa[2]);
    z[3] = qcb[3] * __cosf(xq0.w + qa[3]);
    z[4] = qcb[4] * __cosf(xq1.x + qa[4]);
    z[5] = qcb[5] * __cosf(xq1.y + qa[5]);
    z[6] = qcb[6] * __cosf(xq1.z + qa[6]);
    z[7] = qcb[7] * __cosf(xq1.w + qa[7]);
    float pref[8];
    pref[0] = z[0];
    #pragma unroll
    for (int k = 1; k < 8; ++k) pref[k] = pref[k - 1] * z[k];
    const float q0 = z[1]*z[2]*z[3]*z[4]*z[5]*z[6]*z[7];

    // vectorized gate / hx reads (all 16B-aligned with GSTR/CSTR = 36)
    const float4 fv = *(const float4*)&gate[row * GSTR + h4];
    const float4 iv = *(const float4*)&gate[row * GSTR + 8 + h4];
    const float4 gv = *(const float4*)&gate[row * GSTR + 16 + h4];
    const float4 ov = *(const float4*)&gate[row * GSTR + 24 + h4];
    const float4 hold = *(const float4*)&comb[row * CSTR + 16 + h4];

    const float fA[4] = {fv.x, fv.y, fv.z, fv.w};
    const float iA[4] = {iv.x, iv.y, iv.z, iv.w};
    const float gA[4] = {gv.x, gv.y, gv.z, gv.w};
    const float oA[4] = {ov.x, ov.y, ov.z, ov.w};
    const float hA[4] = {hold.x, hold.y, hold.z, hold.w};

    float4 hn;
    float hnA[4];
    #pragma unroll
    for (int j = 0; j < 4; ++j) {
      // compile-time j, single runtime select on hi (no cmp ladders)
      const float qv = hi ? pref[j + 4] : (j ? pref[j] : q0);
      const float fg = fsigmoid(fA[j]);
      const float ig = fsigmoid(iA[j]);
      const float gg = ftanh(gA[j]);
      const float og = fsigmoid(oA[j]);
      const float cn = fg * cx[j] + ig * (gg + qv);
      cx[j] = cn;
      hnA[j] = og * ftanh(cn) + 0.1f * hA[j];
    }
    hn.x = hnA[0]; hn.y = hnA[1]; hn.z = hnA[2]; hn.w = hnA[3];
    *(float4*)&comb[row * CSTR + 16 + h4] = hn;
    __syncthreads();

    // ---- coalesced hx store: 512B contiguous per wave per step ----
    const int ms = lane >> 1, g4 = (lane & 1) << 2;
    const float4 hv = *(const float4*)&comb[ms * CSTR + 16 + g4];
    *(float4*)(out + ((size_t)s * BATCH + b0 + ms) * HID + g4) = hv;
  }

  // ---- final hx / cx ----
  {
    const size_t HX_OFF = (size_t)S_LEN * BATCH * HID;
    const size_t CX_OFF = HX_OFF + (size_t)BATCH * HID;
    const int ms = lane >> 1, g4 = (lane & 1) << 2;
    const float4 hv = *(const float4*)&comb[ms * CSTR + 16 + g4];
    *(float4*)(out + HX_OFF + ((size_t)b0 + ms) * HID + g4) = hv;
    float4 cv; cv.x = cx[0]; cv.y = cx[1]; cv.z = cx[2]; cv.w = cx[3];
    *(float4*)(out + CX_OFF + ((size_t)b0 + row) * HID + h4) = cv;
  }
}

extern "C" void kernel_launch(void* const* d_in, const int* in_sizes, int n_in,
                              void* d_out, int out_size, void* d_ws, size_t ws_size,
                              hipStream_t stream) {
  (void)in_sizes; (void)n_in; (void)out_size; (void)d_ws; (void)ws_size;
  const float* X        = (const float*)d_in[0];
  const float* conv_w   = (const float*)d_in[1];
  const float* conv_b   = (const float*)d_in[2];
  const float* bn_gamma = (const float*)d_in[3];
  const float* bn_beta  = (const float*)d_in[4];
  const float* bn_mean  = (const float*)d_in[5];
  const float* bn_var   = (const float*)d_in[6];
  const float* w_f      = (const float*)d_in[7];
  const float* b_f      = (const float*)d_in[8];
  const float* w_i      = (const float*)d_in[9];
  const float* b_i      = (const float*)d_in[10];
  const float* w_g      = (const float*)d_in[11];
  const float* b_g      = (const float*)d_in[12];
  const float* w_o      = (const float*)d_in[13];
  const float* b_o      = (const float*)d_in[14];
  const float* q_params = (const float*)d_in[15];
  float* out = (float*)d_out;

  qlstm_scan_kernel<<<dim3(BATCH / 16), dim3(32), 0, stream>>>(
      X, conv_w, conv_b, bn_gamma, bn_beta, bn_mean, bn_var,
      w_f, b_f, w_i, b_i, w_g, b_g, w_o, b_o, q_params, out);
}